// EntityTower_41360535060598
// MI455X (gfx1250) — compile-verified
//
#include <hip/hip_runtime.h>
#include <hip/hip_bf16.h>

#define DFEAT 128
#define NNODES 100000
#define NEDGES 1600000

typedef __attribute__((ext_vector_type(2))) float v2f;
typedef __attribute__((ext_vector_type(8))) float v8f;

// ---------------------------------------------------------------------------
// deg[i] = 1.0 (self loop weight)
__global__ void gcn_deg_init(float* __restrict__ deg, int n) {
    int i = blockIdx.x * blockDim.x + threadIdx.x;
    if (i < n) deg[i] = 1.0f;
}

// deg[dst[e]] += ew[e]
__global__ void gcn_deg_accum(const int* __restrict__ dst, const float* __restrict__ ew,
                              float* __restrict__ deg, int e) {
    int i = blockIdx.x * blockDim.x + threadIdx.x;
    if (i < e) atomicAdd(&deg[dst[i]], ew[i]);
}

// deg -> dinv = rsqrt(deg) (deg >= 1 here, keep guard for fidelity)
__global__ void gcn_dinv(float* __restrict__ deg, int n) {
    int i = blockIdx.x * blockDim.x + threadIdx.x;
    if (i < n) {
        float d = deg[i];
        deg[i] = (d > 0.0f) ? __frsqrt_rn(d) : 0.0f;
    }
}

// ---------------------------------------------------------------------------
// h[nrows,128] = A[nrows,128] @ W[128,128]  via V_WMMA_F32_16X16X4_F32.
// One wave -> 16 rows x 128 cols (8 accumulator tiles). W staged in LDS.
__global__ void __launch_bounds__(256)
gcn_gemm_f32_wmma(const float* __restrict__ A, const float* __restrict__ W,
                  float* __restrict__ out, int nrows) {
    __shared__ float lW[DFEAT * DFEAT];           // 64 KB of the 320 KB WGP LDS
    for (int i = threadIdx.x; i < DFEAT * DFEAT; i += blockDim.x)
        lW[i] = W[i];
    __syncthreads();

    const int wave = threadIdx.x >> 5;            // 8 waves / block
    const int lane = threadIdx.x & 31;
    const int half = lane >> 4;                   // 0: K=k,k+1  1: K=k+2,k+3
    const int l16  = lane & 15;

    const int row0 = (blockIdx.x * 8 + wave) * 16;
    if (row0 >= nrows) return;                    // wave-uniform: EXEC stays all-1s

    v8f acc[8] = {};                              // 16x128 f32 strip, 64 VGPRs

    const float* arow = A + (size_t)(row0 + l16) * DFEAT;
    for (int k = 0; k < DFEAT; k += 4) {
        const int ka = k + half * 2;
        v2f af;                                   // A 16x4 frag per ISA layout
        af.x = arow[ka];
        af.y = arow[ka + 1];
#pragma unroll
        for (int ct = 0; ct < 8; ++ct) {
            const int col = ct * 16 + l16;
            v2f bf;                               // B 4x16 frag (row striped across lanes)
            bf.x = lW[ka * DFEAT + col];
            bf.y = lW[(ka + 1) * DFEAT + col];
            acc[ct] = __builtin_amdgcn_wmma_f32_16x16x4_f32(
                false, af, false, bf, (short)0, acc[ct], false, false);
        }
    }

    // C/D layout: lanes 0-15 hold rows row0+j, lanes 16-31 rows row0+8+j
    const int rbase = row0 + half * 8;
#pragma unroll
    for (int ct = 0; ct < 8; ++ct) {
        const int col = ct * 16 + l16;
#pragma unroll
        for (int j = 0; j < 8; ++j)
            out[(size_t)(rbase + j) * DFEAT + col] = acc[ct][j];
    }
}

// ---------------------------------------------------------------------------
// out[i,f] = b[f] + dinv[i]^2 * h[i,f]   (bias + self-loop message)
__global__ void gcn_init_out(float* __restrict__ out, const float* __restrict__ h,
                             const float* __restrict__ dinv, const float* __restrict__ b,
                             int total) {
    int i = blockIdx.x * blockDim.x + threadIdx.x;
    if (i >= total) return;
    const int node = i >> 7;
    const int f    = i & (DFEAT - 1);
    const float di = dinv[node];
    out[i] = b[f] + di * di * h[i];
}

// One wave per edge: out[dst,:] += norm * h[src,:]  (float4 lanes, f32 atomics)
__global__ void __launch_bounds__(256)
gcn_edge_scatter(const float* __restrict__ h, const int* __restrict__ src,
                 const int* __restrict__ dst, const float* __restrict__ ew,
                 const float* __restrict__ dinv, float* __restrict__ out, int e) {
    const int edge = blockIdx.x * (blockDim.x >> 5) + (threadIdx.x >> 5);
    if (edge >= e) return;
    const int lane = threadIdx.x & 31;

    const int s = src[edge];                      // wave-uniform -> scalar loads
    const int d = dst[edge];
    const float norm = dinv[s] * ew[edge] * dinv[d];

    const float4 v = ((const float4*)(h + (size_t)s * DFEAT))[lane];  // coalesced 512B, L2-hot
    float* o = out + (size_t)d * DFEAT + lane * 4;
    atomicAdd(o + 0, v.x * norm);
    atomicAdd(o + 1, v.y * norm);
    atomicAdd(o + 2, v.z * norm);
    atomicAdd(o + 3, v.w * norm);
}

__global__ void gcn_relu(float* __restrict__ x, int total) {
    int i = blockIdx.x * blockDim.x + threadIdx.x;
    if (i < total) x[i] = fmaxf(x[i], 0.0f);
}

// ---------------------------------------------------------------------------
extern "C" void kernel_launch(void* const* d_in, const int* in_sizes, int n_in,
                              void* d_out, int out_size, void* d_ws, size_t ws_size,
                              hipStream_t stream) {
    const float* x   = (const float*)d_in[0];
    const int*   ei  = (const int*)d_in[1];      // [2, E] int32 (JAX x64 off)
    const float* ew  = (const float*)d_in[2];
    const float* W1  = (const float*)d_in[3];
    const float* b1  = (const float*)d_in[4];
    const float* W2  = (const float*)d_in[5];
    const float* b2  = (const float*)d_in[6];
    float* out = (float*)d_out;

    const int N = NNODES, E = NEDGES;
    const int src_off = 0, dst_off = E;
    const int* src = ei + src_off;
    const int* dst = ei + dst_off;

    float* h    = (float*)d_ws;                  // N*D
    float* y1   = h + (size_t)N * DFEAT;         // N*D
    float* dinv = y1 + (size_t)N * DFEAT;        // N

    const int totalND = N * DFEAT;
    const int B = 256;

    // Degree normalization (shared by both layers)
    gcn_deg_init<<<(N + B - 1) / B, B, 0, stream>>>(dinv, N);
    gcn_deg_accum<<<(E + B - 1) / B, B, 0, stream>>>(dst, ew, dinv, E);
    gcn_dinv<<<(N + B - 1) / B, B, 0, stream>>>(dinv, N);

    const int gemmBlocks = ((N / 16) + 7) / 8;   // 8 row-tiles (waves) per block
    const int edgeBlocks = (E + 7) / 8;          // 8 edges (waves) per block

    // Layer 1
    gcn_gemm_f32_wmma<<<gemmBlocks, B, 0, stream>>>(x, W1, h, N);
    gcn_init_out<<<(totalND + B - 1) / B, B, 0, stream>>>(y1, h, dinv, b1, totalND);
    gcn_edge_scatter<<<edgeBlocks, B, 0, stream>>>(h, src, dst, ew, dinv, y1, E);
    gcn_relu<<<(totalND + B - 1) / B, B, 0, stream>>>(y1, totalND);

    // Layer 2
    gcn_gemm_f32_wmma<<<gemmBlocks, B, 0, stream>>>(y1, W2, h, N);
    gcn_init_out<<<(totalND + B - 1) / B, B, 0, stream>>>(out, h, dinv, b2, totalND);
    gcn_edge_scatter<<<edgeBlocks, B, 0, stream>>>(h, src, dst, ew, dinv, out, E);
    gcn_relu<<<(totalND + B - 1) / B, B, 0, stream>>>(out, totalND);
}